// InputDefenseLayer_31920196944077
// MI455X (gfx1250) — compile-verified
//
#include <hip/hip_runtime.h>

typedef float v2f __attribute__((ext_vector_type(2)));
typedef float v8f __attribute__((ext_vector_type(8)));

#define BATCH   64
#define TSTEPS  2048
#define FEAT    256
#define TILE_T  16
#define TILE_F  16
#define N_TILES (TSTEPS / TILE_T)   /* 128 */
#define WAVES_PER_BLOCK 4

__device__ __forceinline__ float clip35(float v) {
    return fminf(fmaxf(v, -3.5f), 3.5f);
}

// EMA scan s_t = 0.25*x_t + 0.75*s_{t-1}, s_0 = x_0 (after clip), expressed as
// blocked matrix recurrence: per 16-step tile, D = L * X + C, where
// L[i][j] = 0.25*0.75^(i-j) (lower triangular) and C[m][n] = 0.75^(m+1)*carry[n].
// L*X is computed with 4 chained V_WMMA_F32_16X16X4_F32 (K=4 each).
// Streaming workload (no reuse, 268 MB > 192 MB L2) -> NT load/store hints.
__global__ __launch_bounds__(32 * WAVES_PER_BLOCK)
void ema_scan_wmma_31920196944077(const float* __restrict__ x,
                                  float* __restrict__ out) {
    const int lane = threadIdx.x & 31;
    const int wave = threadIdx.x >> 5;
    const int hi   = lane >> 4;     // 0: lanes 0-15, 1: lanes 16-31
    const int n    = lane & 15;     // feature-in-tile (B/C/D col), also A row m
    const int b    = blockIdx.y;
    const int f0   = (blockIdx.x * WAVES_PER_BLOCK + wave) * TILE_F;

    const float* colIn  = x   + (size_t)b * TSTEPS * FEAT + f0 + n;
    float*       colOut = out + (size_t)b * TSTEPS * FEAT + f0 + n;

    // Powers of the decay 0.75
    float p75[TILE_T + 1];
    p75[0] = 1.0f;
    #pragma unroll
    for (int i = 1; i <= TILE_T; ++i) p75[i] = p75[i - 1] * 0.75f;

    // A-matrix chunks (16x4 f32 layout: lane m<16 holds K=2*hi+v in VGPR v).
    // Chunk kc covers global K = 4*kc .. 4*kc+3.
    const int m = n;
    v2f a[4];
    #pragma unroll
    for (int kc = 0; kc < 4; ++kc) {
        #pragma unroll
        for (int v = 0; v < 2; ++v) {
            const int j = 4 * kc + 2 * hi + v;
            a[kc][v] = (j <= m) ? (0.25f * p75[m - j]) : 0.0f;
        }
    }

    // C/D VGPR r holds row m' = r + 8*hi; carry coefficient 0.75^(m'+1)
    float coef[8];
    #pragma unroll
    for (int r = 0; r < 8; ++r) coef[r] = p75[r + 8 * hi + 1];

    // B layout (4x16 f32 per chunk, 2 VGPRs): reg r=2*kc+v holds row K=2*r+hi.
    // Preload tile 0; fold s_0 = clip(x_0) by scaling the K=0 element by 1/alpha.
    float bcur[8];
    #pragma unroll
    for (int r = 0; r < 8; ++r)
        bcur[r] = clip35(__builtin_nontemporal_load(
                             &colIn[(size_t)(2 * r + hi) * FEAT]));
    if (hi == 0) bcur[0] *= 4.0f;

    float carry = 0.0f;

    for (int tile = 0; tile < N_TILES; ++tile) {
        const int t0 = tile * TILE_T;

        // Software pipeline: issue next tile's loads before this tile's WMMAs.
        float bnext[8];
        if (tile + 1 < N_TILES) {
            #pragma unroll
            for (int r = 0; r < 8; ++r)
                bnext[r] = clip35(__builtin_nontemporal_load(
                    &colIn[(size_t)(t0 + TILE_T + 2 * r + hi) * FEAT]));
        }
        if (tile + 2 < N_TILES) {
            __builtin_prefetch(&colIn[(size_t)(t0 + 2 * TILE_T) * FEAT], 0, 0);
        }

        // C operand = incoming-carry contribution
        v8f d;
        #pragma unroll
        for (int r = 0; r < 8; ++r) d[r] = coef[r] * carry;

        // D = L * X + C via 4 chained 16x16x4 f32 WMMAs
        #pragma unroll
        for (int kc = 0; kc < 4; ++kc) {
            v2f bb;
            bb[0] = bcur[2 * kc];
            bb[1] = bcur[2 * kc + 1];
            d = __builtin_amdgcn_wmma_f32_16x16x4_f32(
                    false, a[kc], false, bb, (short)0, d, false, false);
        }

        // New carry = D row 15 (VGPR 7, lanes 16-31) broadcast to both halves
        carry = __shfl(d[7], 16 + n, 32);

        #pragma unroll
        for (int r = 0; r < 8; ++r)
            __builtin_nontemporal_store(
                d[r], &colOut[(size_t)(t0 + r + 8 * hi) * FEAT]);

        if (tile + 1 < N_TILES) {
            #pragma unroll
            for (int r = 0; r < 8; ++r) bcur[r] = bnext[r];
        }
    }
}

extern "C" void kernel_launch(void* const* d_in, const int* in_sizes, int n_in,
                              void* d_out, int out_size, void* d_ws, size_t ws_size,
                              hipStream_t stream) {
    (void)in_sizes; (void)n_in; (void)d_ws; (void)ws_size; (void)out_size;
    const float* x = (const float*)d_in[0];
    float* out = (float*)d_out;

    dim3 grid(FEAT / (TILE_F * WAVES_PER_BLOCK), BATCH);  // (4, 64)
    dim3 block(32 * WAVES_PER_BLOCK);                     // 128 threads = 4 waves
    ema_scan_wmma_31920196944077<<<grid, block, 0, stream>>>(x, out);
}